// FeatureNet_83193516523979
// MI455X (gfx1250) — compile-verified
//
#include <hip/hip_runtime.h>
#include <cstdint>
#include <cstddef>

typedef __attribute__((ext_vector_type(2))) float v2f;
typedef __attribute__((ext_vector_type(4))) float v4f;
typedef __attribute__((ext_vector_type(8))) float v8f;

#define K_VOX 12000
#define T_PTS 35
#define F_IN  7
#define DD 10
#define HH 400
#define WW 352
#define COUT 128
#define TP 48               // T padded to 3 WMMA M-tiles
#define XP_STR 9            // padded LDS row stride for xpad [48][8]
#define X2_STR 33           // padded LDS row stride for x2   [48][32]
#define LDS_PER_WAVE (TP*XP_STR + TP*X2_STR + TP)   // 432 + 1584 + 48 = 2064 floats
#define WAVES 4
#define EPS 1e-3f

// ---------------------------------------------------------------------------
// Kernel 0: zero the 721MB output grid with vectorized non-temporal stores.
// This is the HBM-bound part of the problem (~31us floor at 23.3 TB/s).
// ---------------------------------------------------------------------------
__global__ void zero_out_kernel(v4f* __restrict__ p, size_t n4) {
    size_t i = (size_t)blockIdx.x * blockDim.x + threadIdx.x;
    size_t stride = (size_t)gridDim.x * blockDim.x;
    v4f z = {0.f, 0.f, 0.f, 0.f};
    for (; i < n4; i += stride)
        __builtin_nontemporal_store(z, &p[i]);
}

// ---------------------------------------------------------------------------
// Kernel 1: one wave32 per voxel. Full VFE chain with V_WMMA_F32_16X16X4_F32,
// LDS staging for the D-layout -> A-layout transpose between layers, then
// global_atomic_add_f32 scatter (duplicate coords accumulate, matching
// tf.scatter_nd semantics).
// ---------------------------------------------------------------------------
__global__ void __launch_bounds__(WAVES * 32)
vfe_scatter_kernel(const float* __restrict__ feat,
                   const float* __restrict__ w1,  const float* __restrict__ b1,
                   const float* __restrict__ g1,  const float* __restrict__ be1,
                   const float* __restrict__ mu1, const float* __restrict__ va1,
                   const float* __restrict__ w2,  const float* __restrict__ b2,
                   const float* __restrict__ g2,  const float* __restrict__ be2,
                   const float* __restrict__ mu2, const float* __restrict__ va2,
                   const int*   __restrict__ coord,
                   float*       __restrict__ out)
{
    __shared__ float smem[WAVES * LDS_PER_WAVE];

    const int lane = threadIdx.x & 31;
    const int wv   = threadIdx.x >> 5;
    const int vox  = blockIdx.x * WAVES + wv;   // grid exactly covers K_VOX

    float* xpad = smem + wv * LDS_PER_WAVE;     // [48][XP_STR], features padded [48][8]
    float* x2   = xpad + TP * XP_STR;           // [48][X2_STR], layer-2 input [48][32]
    float* msk  = x2   + TP * X2_STR;           // [48] point-valid mask

    const int n  = lane & 15;   // WMMA column (output channel within N-tile)
    const int kh = lane >> 4;   // K-half (A/B layout) == M-half (C/D layout)

    // ---- Phase A: stage features into LDS (zero-padded), compute mask -----
    const float* fv = feat + (size_t)vox * (T_PTS * F_IN);
    for (int idx = lane; idx < TP * 8; idx += 32) {
        int m = idx >> 3, k = idx & 7;
        float v = 0.f;
        if (m < T_PTS && k < F_IN) v = fv[m * F_IN + k];
        xpad[m * XP_STR + k] = v;
    }
    for (int m = lane; m < TP; m += 32) {
        float mx = -__builtin_inff();
        if (m < T_PTS)
            for (int k = 0; k < F_IN; ++k) mx = fmaxf(mx, fv[m * F_IN + k]);
        msk[m] = (m < T_PTS && mx != 0.f) ? 1.f : 0.f;
    }
    __asm__ volatile("s_wait_dscnt 0x0" ::: "memory");

    float vcnt = 0.f;
    for (int m = 0; m < T_PTS; ++m) vcnt += msk[m];   // broadcast LDS reads
    const bool anyInvalid = (vcnt < (float)T_PTS);

    // ---- Phase B: layer 1 WMMA (K=8 padded, N=16), bias+ReLU+BN ------------
    const float bb1 = b1[n], gg1 = g1[n], bt1 = be1[n], mm1 = mu1[n];
    const float rs1 = rsqrtf(va1[n] + EPS);

    v2f B1[2];
    for (int s = 0; s < 2; ++s) {
        int k0 = 4 * s + 2 * kh;
        B1[s].x = (k0     < F_IN) ? w1[ k0      * 16 + n] : 0.f;
        B1[s].y = (k0 + 1 < F_IN) ? w1[(k0 + 1) * 16 + n] : 0.f;
    }

    v8f pw[3];
    for (int t = 0; t < 3; ++t) {
        v8f acc = {};
        const int ma = n + 16 * t;            // A row for this lane
        for (int s = 0; s < 2; ++s) {
            v2f a;
            a.x = xpad[ma * XP_STR + 4 * s + 2 * kh];
            a.y = xpad[ma * XP_STR + 4 * s + 2 * kh + 1];
            acc = __builtin_amdgcn_wmma_f32_16x16x4_f32(
                      false, a, false, B1[s], (short)0, acc, false, false);
        }
        for (int i = 0; i < 8; ++i) {
            float d = fmaxf(acc[i] + bb1, 0.f);
            pw[t][i] = gg1 * (d - mm1) * rs1 + bt1;
        }
    }

    // agg1 = max over all 35 rows of pw1 (unmasked, matches reference)
    float agg1 = -__builtin_inff();
    for (int t = 0; t < 3; ++t)
        for (int i = 0; i < 8; ++i) {
            int r = i + 8 * kh + 16 * t;      // D-layout row
            agg1 = fmaxf(agg1, (r < T_PTS) ? pw[t][i] : -__builtin_inff());
        }
    agg1 = fmaxf(agg1, __shfl_xor(agg1, 16, 32));

    // ---- Phase D: write masked concat [pw1, agg1] to LDS in row-major ------
    for (int t = 0; t < 3; ++t)
        for (int i = 0; i < 8; ++i) {
            int r = i + 8 * kh + 16 * t;
            float mk = msk[r];                // 0 for r >= 35
            x2[r * X2_STR + n]      = mk * pw[t][i];
            x2[r * X2_STR + 16 + n] = mk * agg1;
        }
    __asm__ volatile("s_wait_dscnt 0x0" ::: "memory");

    // ---- Phase E: layer 2 WMMA (K=32, N=64 in 4 tiles) + scatter -----------
    const int* cr = coord + (size_t)vox * 4;
    const size_t obase =
        ((((size_t)cr[0] * DD + cr[1]) * HH + cr[2]) * WW + cr[3]) * COUT;

    for (int nt = 0; nt < 4; ++nt) {
        const int c  = nt * 16 + n;
        const float bb = b2[c], gg = g2[c], bt = be2[c], mm = mu2[c];
        const float rs = rsqrtf(va2[c] + EPS);

        v2f Br[8];
        for (int s = 0; s < 8; ++s) {
            int k0 = 4 * s + 2 * kh;
            Br[s].x = w2[ k0      * 64 + c];
            Br[s].y = w2[(k0 + 1) * 64 + c];
        }

        float vmax_mk = -__builtin_inff();    // max over rows of mask*pw2
        float vmax_un = -__builtin_inff();    // max over rows of pw2 (agg2)
        for (int t = 0; t < 3; ++t) {
            v8f acc = {};
            const int ma = n + 16 * t;
            for (int s = 0; s < 8; ++s) {
                v2f a;
                a.x = x2[ma * X2_STR + 4 * s + 2 * kh];
                a.y = x2[ma * X2_STR + 4 * s + 2 * kh + 1];
                acc = __builtin_amdgcn_wmma_f32_16x16x4_f32(
                          false, a, false, Br[s], (short)0, acc, false, false);
            }
            for (int i = 0; i < 8; ++i) {
                int r = i + 8 * kh + 16 * t;
                float d  = fmaxf(acc[i] + bb, 0.f);
                float p  = gg * (d - mm) * rs + bt;
                bool  vr = (r < T_PTS);
                float mk = msk[r];
                vmax_un = fmaxf(vmax_un, vr ? p      : -__builtin_inff());
                vmax_mk = fmaxf(vmax_mk, vr ? mk * p : -__builtin_inff());
            }
        }
        vmax_un = fmaxf(vmax_un, __shfl_xor(vmax_un, 16, 32));
        vmax_mk = fmaxf(vmax_mk, __shfl_xor(vmax_mk, 16, 32));
        // concat-agg channel: masked rows contribute 0 when any pad row exists
        float vagg = anyInvalid ? fmaxf(vmax_un, 0.f) : vmax_un;

        // lanes 0-15: pw channel c ; lanes 16-31: agg channel 64+c
        float val = (kh == 0) ? vmax_mk : vagg;
        int   ch  = (kh == 0) ? c : (64 + c);
        atomicAdd(out + obase + ch, val);
    }
}

// ---------------------------------------------------------------------------
extern "C" void kernel_launch(void* const* d_in, const int* in_sizes, int n_in,
                              void* d_out, int out_size, void* d_ws, size_t ws_size,
                              hipStream_t stream) {
    (void)in_sizes; (void)n_in; (void)d_ws; (void)ws_size;

    const float* feat = (const float*)d_in[0];
    const float* w1   = (const float*)d_in[1];
    const float* b1   = (const float*)d_in[2];
    const float* g1   = (const float*)d_in[3];
    const float* be1  = (const float*)d_in[4];
    const float* mu1  = (const float*)d_in[5];
    const float* va1  = (const float*)d_in[6];
    const float* w2   = (const float*)d_in[7];
    const float* b2   = (const float*)d_in[8];
    const float* g2   = (const float*)d_in[9];
    const float* be2  = (const float*)d_in[10];
    const float* mu2  = (const float*)d_in[11];
    const float* va2  = (const float*)d_in[12];
    const int*   crd  = (const int*)d_in[13];
    float* out = (float*)d_out;

    // 1) zero-fill the scatter grid (HBM-bound phase)
    size_t n4 = (size_t)out_size / 4;                 // out_size divisible by 4
    int zblocks = (int)((n4 + 255) / 256);
    zero_out_kernel<<<zblocks, 256, 0, stream>>>((v4f*)d_out, n4);

    // 2) WMMA VFE + atomic scatter, one wave per voxel
    vfe_scatter_kernel<<<K_VOX / WAVES, WAVES * 32, 0, stream>>>(
        feat, w1, b1, g1, be1, mu1, va1,
        w2, b2, g2, be2, mu2, va2, crd, out);
}